// Model21H_31731218382997
// MI455X (gfx1250) — compile-verified
//
#include <hip/hip_runtime.h>
#include <hip/hip_bf16.h>
#include <math.h>

// ---------------------------------------------------------------------------
// Icosphere U-Net forward for MI455X (gfx1250, wave32).
// - All GEMMs (gather-convs and attention matmuls) on V_WMMA_F32_16X16X4_F32
//   (fp32 accumulate, exact-precision match to the fp32 reference).
// - Up to 4 waves/block share one LDS-staged gathered A tile; each wave owns a
//   16-wide output-channel slice (A gather traffic cut 4x on wide layers).
// - Staging uses GLOBAL_LOAD_ASYNC_TO_LDS_B32 (ASYNCcnt) when the toolchain
//   exposes the builtin: memory -> LDS with no VGPR round trip.
// ---------------------------------------------------------------------------

typedef __attribute__((ext_vector_type(2))) float v2f;
typedef __attribute__((ext_vector_type(8))) float v8f;

#define CT 64           // K-chunk staged in LDS per step
#define LDSP (CT + 1)   // padded row stride (conflict-free column reads)

#if defined(__has_builtin)
#  if __has_builtin(__builtin_amdgcn_global_load_async_to_lds_b32)
#    define HAVE_ASYNC_LDS 1
#  endif
#endif
#ifndef HAVE_ASYNC_LDS
#  define HAVE_ASYNC_LDS 0
#endif

typedef __attribute__((address_space(1))) int g_i32;
typedef __attribute__((address_space(3))) int l_i32;

static __device__ __forceinline__ void stage_one(float* dst, const float* src)
{
#if HAVE_ASYNC_LDS
    float* s  = const_cast<float*>(src);
    g_i32* gp = (g_i32*)s;
    l_i32* lp = (l_i32*)dst;
    __builtin_amdgcn_global_load_async_to_lds_b32(gp, lp, 0, 0);
#else
    *dst = *src;
#endif
}

static __device__ __forceinline__ void wait_async_lds()
{
#if HAVE_ASYNC_LDS
#  if defined(__has_builtin) && __has_builtin(__builtin_amdgcn_s_wait_asynccnt)
    __builtin_amdgcn_s_wait_asynccnt(0);
#  else
    asm volatile("s_wait_asynccnt 0" ::: "memory");
#  endif
#endif
}

// out[bt,n,o] = act( scale? * ( sum_{k<K,c<Ci} in[bt, neigh[n,k], c] * W[k,c,o] ) + b[o] )
// neigh == NULL  -> identity gather (dense GEMM path for attention), use K=1.
// scale2 != NULL -> multiply row (bt,n) by scale2[(bt*N+n)*2 + scaleSel].
// Grid: x = o-tile groups (blockDim/32 tiles of 16 each), y = bt, z = n-tiles.
__global__ __launch_bounds__(128)
void sconv_wmma(const float* __restrict__ in, const int* __restrict__ neigh, int K,
                const float* __restrict__ W, const float* __restrict__ bias,
                const float* __restrict__ scale2, int scaleSel,
                float* __restrict__ out, int N, int Ci, int Co, int relu)
{
    __shared__ float As[16 * LDSP];

    const int tid     = threadIdx.x;
    const int lane    = tid & 31;
    const int waveId  = tid >> 5;
    const int wavesPB = blockDim.x >> 5;
    const int row     = lane & 15;
    const int half    = lane >> 4;

    const int n0 = blockIdx.z << 4;
    const int o0 = (blockIdx.x * wavesPB + waveId) << 4;
    const int bt = blockIdx.y;

    v8f acc = {0.f, 0.f, 0.f, 0.f, 0.f, 0.f, 0.f, 0.f};

    const size_t inBase = (size_t)bt * N * Ci;
    const int o   = o0 + row;
    const bool ov = (o < Co);

    for (int k = 0; k < K; ++k) {
        const float* Wk = W + (size_t)k * Ci * Co;
        for (int cc0 = 0; cc0 < Ci; cc0 += CT) {
            int ce = Ci - cc0; if (ce > CT) ce = CT;
            int jpad = (ce + 3) & ~3;
            // --- cooperative stage of gathered rows into LDS (async -> LDS) ---
            for (int t = tid; t < 16 * jpad; t += blockDim.x) {
                int m, j;
                if (jpad == CT) { m = t >> 6; j = t & 63; }
                else           { m = t / jpad; j = t - m * jpad; }
                int nn = n0 + m; if (nn >= N) nn = N - 1;     // clamp; store is masked
                int v = neigh ? neigh[nn * K + k] : nn;
                if (j < ce) stage_one(&As[m * LDSP + j], &in[inBase + (size_t)v * Ci + (cc0 + j)]);
                else        As[m * LDSP + j] = 0.0f;
            }
            wait_async_lds();
            __syncthreads();
            // --- WMMA K-loop: A 16x4 (lane<16: K0,K1 ; lane+16: K2,K3), B 4x16 ---
            const int steps = jpad >> 2;
            const float* As_r = &As[row * LDSP];
            #pragma unroll 4
            for (int jj = 0; jj < steps; ++jj) {
                v2f a, b;
                int ja = (jj << 2) + (half << 1);
                a.x = As_r[ja];
                a.y = As_r[ja + 1];
                int cb = cc0 + ja;
                b.x = (ov && (cb     < Ci)) ? Wk[(size_t)cb       * Co + o] : 0.0f;
                b.y = (ov && (cb + 1 < Ci)) ? Wk[(size_t)(cb + 1) * Co + o] : 0.0f;
                acc = __builtin_amdgcn_wmma_f32_16x16x4_f32(
                          false, a, false, b, (short)0, acc, false, false);
            }
            __syncthreads();
        }
    }

    // --- store: VGPR r holds M = r (lanes 0-15) / M = 8+r (lanes 16-31), N = lane&15 ---
    if (ov) {
        float bv = bias ? bias[o] : 0.0f;
        for (int r = 0; r < 8; ++r) {
            int m = r + (half << 3);
            int n = n0 + m;
            if (n < N) {
                float v = acc[r] + bv;
                if (scale2) v = acc[r] * scale2[((size_t)bt * N + n) * 2 + scaleSel];
                if (relu) v = fmaxf(v, 0.0f);
                out[((size_t)bt * N + n) * Co + o] = v;
            }
        }
    }
}

// out[bt,n,c] = max_{k<7} in[bt, neigh[n,k], c],  n < Nout
__global__ void spool_max(const float* __restrict__ in, const int* __restrict__ neigh,
                          float* __restrict__ out, int Nin, int Nout, int C)
{
    size_t idx = (size_t)blockIdx.x * blockDim.x + threadIdx.x;
    size_t total = (size_t)16 * Nout * C;
    if (idx >= total) return;
    int c = (int)(idx % C);
    size_t t = idx / C;
    int n  = (int)(t % Nout);
    int bt = (int)(t / Nout);
    float m = -3.4e38f;
    for (int k = 0; k < 7; ++k) {
        int v = neigh[n * 7 + k];
        m = fmaxf(m, in[((size_t)bt * Nin + v) * C + c]);
    }
    out[((size_t)bt * Nout + n) * C + c] = m;
}

// per-vertex 2-way softmax over modality scores; one wave32 per (bt,n) row
__global__ __launch_bounds__(256)
void att_score(const float* __restrict__ f, const float* __restrict__ a,
               const float* __restrict__ wf, const float* __restrict__ wa,
               float* __restrict__ w2, int N, int C)
{
    int wid  = (int)(((size_t)blockIdx.x * blockDim.x + threadIdx.x) >> 5);
    int lane = threadIdx.x & 31;
    if (wid >= 16 * N) return;                     // wave-uniform exit
    const float* fr = f + (size_t)wid * C;
    const float* ar = a + (size_t)wid * C;
    float sf = 0.f, sa = 0.f;
    for (int c = lane; c < C; c += 32) { sf += fr[c] * wf[c]; sa += ar[c] * wa[c]; }
    for (int s = 16; s; s >>= 1) {
        sf += __shfl_xor(sf, s, 32);
        sa += __shfl_xor(sa, s, 32);
    }
    if (lane == 0) {
        float mx = fmaxf(sf, sa);
        float ef = expf(sf - mx), ea = expf(sa - mx);
        float inv = 1.0f / (ef + ea);
        w2[(size_t)wid * 2 + 0] = ef * inv;
        w2[(size_t)wid * 2 + 1] = ea * inv;
    }
}

// out[bt,n, :] = concat of up-to-4 sources; sources with u?=1 are gathered via up[n]
__global__ void concat_gather(float* __restrict__ out, int Nout, int Ctot,
                              const int* __restrict__ up,
                              const float* s0, int c0, int n0s, int u0,
                              const float* s1, int c1, int n1s, int u1,
                              const float* s2, int c2, int n2s, int u2,
                              const float* s3, int c3, int n3s, int u3)
{
    size_t idx = (size_t)blockIdx.x * blockDim.x + threadIdx.x;
    size_t total = (size_t)16 * Nout * Ctot;
    if (idx >= total) return;
    int c = (int)(idx % Ctot);
    size_t t = idx / Ctot;
    int n  = (int)(t % Nout);
    int bt = (int)(t / Nout);
    const float* s; int Cs, sN, uu, cc;
    if (c < c0)                { s = s0; Cs = c0; sN = n0s; uu = u0; cc = c; }
    else if (c < c0 + c1)      { s = s1; Cs = c1; sN = n1s; uu = u1; cc = c - c0; }
    else if (c < c0 + c1 + c2) { s = s2; Cs = c2; sN = n2s; uu = u2; cc = c - c0 - c1; }
    else                       { s = s3; Cs = c3; sN = n3s; uu = u3; cc = c - c0 - c1 - c2; }
    int r = uu ? up[n] : n;
    out[idx] = s[((size_t)bt * sN + r) * Cs + cc];
}

// row softmax over N vertices, one block per bt
__global__ __launch_bounds__(256)
void softmax_rows(const float* __restrict__ in, float* __restrict__ out, int N)
{
    __shared__ float red[256];
    int bt = blockIdx.x, tid = threadIdx.x;
    const float* row = in + (size_t)bt * N;
    float m = -3.4e38f;
    for (int i = tid; i < N; i += 256) m = fmaxf(m, row[i]);
    red[tid] = m; __syncthreads();
    for (int s = 128; s; s >>= 1) { if (tid < s) red[tid] = fmaxf(red[tid], red[tid + s]); __syncthreads(); }
    m = red[0]; __syncthreads();
    float sum = 0.f;
    for (int i = tid; i < N; i += 256) sum += expf(row[i] - m);
    red[tid] = sum; __syncthreads();
    for (int s = 128; s; s >>= 1) { if (tid < s) red[tid] += red[tid + s]; __syncthreads(); }
    float inv = 1.0f / red[0];
    for (int i = tid; i < N; i += 256) out[(size_t)bt * N + i] = expf(row[i] - m) * inv;
}

// ---------------------------------------------------------------------------
// Host orchestration
// ---------------------------------------------------------------------------

enum {
    IN_FRAME = 0, IN_AEM = 1, IN_TLOC = 2,
    C0F_W = 3, C0F_B, C0A_W, C0A_B, C1F_W, C1F_B, C1A_W, C1A_B,
    C2F_W, C2F_B, C2A_W, C2A_B, C3F_W, C3F_B, C3A_W, C3A_B,
    C4F_W, C4F_B, C4A_W, C4A_B, C5_W, C5_B, C6_W, C6_B,
    C7_W, C7_B, C8_W, C8_B, C9_W, C9_B,
    A0_WF, A0_WA, A0_MF, A0_MA, A1_WF, A1_WA, A1_MF, A1_MA,
    A2_WF, A2_WA, A2_MF, A2_MA, A3_WF, A3_WA, A3_MF, A3_MA,
    A4_WF, A4_WA, A4_MF, A4_MA,
    NEI5, NEI4, NEI3, NEI2, NEI1, UP2, UP3, UP4, UP5, N_INPUTS
};

static const int BT = 16;
static const int N5 = 10242, N4 = 2562, N3 = 642, N2 = 162, N1 = 42;

static inline int cdiv(long long a, int b) { return (int)((a + b - 1) / b); }

static inline void launch_sconv(hipStream_t s, const float* in, const int* neigh, int K,
                                const float* W, const float* b,
                                const float* scale2, int sel,
                                float* out, int N, int Ci, int Co, int relu)
{
    int wavesPB = cdiv(Co, 16); if (wavesPB > 4) wavesPB = 4;
    int nOg = cdiv(Co, 16 * wavesPB);
    dim3 g((unsigned)nOg, BT, (unsigned)cdiv(N, 16));
    sconv_wmma<<<g, 32 * wavesPB, 0, s>>>(in, neigh, K, W, b, scale2, sel, out, N, Ci, Co, relu);
}

extern "C" void kernel_launch(void* const* d_in, const int* in_sizes, int n_in,
                              void* d_out, int out_size, void* d_ws, size_t ws_size,
                              hipStream_t stream)
{
    if (n_in < N_INPUTS) return;
    (void)in_sizes; (void)out_size; (void)ws_size;

    auto F = [&](int i) { return (const float*)d_in[i]; };
    auto I = [&](int i) { return (const int*)d_in[i]; };

    const float* frame = F(IN_FRAME);
    const float* aem   = F(IN_AEM);
    const int* nei5 = I(NEI5); const int* nei4 = I(NEI4); const int* nei3 = I(NEI3);
    const int* nei2 = I(NEI2); const int* nei1 = I(NEI1);
    const int* up2 = I(UP2); const int* up3 = I(UP3); const int* up4 = I(UP4); const int* up5 = I(UP5);

    // bump allocator over workspace
    float* wsp = (float*)d_ws;
    size_t off = 0;
    auto alloc = [&](size_t n) { float* p = wsp + off; off += n; return p; };
    auto elw = [&](size_t total) { return dim3((unsigned)cdiv((long long)total, 256)); };

    // ---------- encoder ----------
    // level 5
    float* c0f = alloc((size_t)BT * N5 * 32);
    float* c0a = alloc((size_t)BT * N5 * 32);
    launch_sconv(stream, frame, nei5, 7, F(C0F_W), F(C0F_B), nullptr, 0, c0f, N5, 3, 32, 1);
    launch_sconv(stream, aem,   nei5, 7, F(C0A_W), F(C0A_B), nullptr, 0, c0a, N5, 1, 32, 1);
    float* s0f = alloc((size_t)BT * N4 * 32);
    float* s0a = alloc((size_t)BT * N4 * 32);
    spool_max<<<elw((size_t)BT * N4 * 32), 256, 0, stream>>>(c0f, nei5, s0f, N5, N4, 32);
    spool_max<<<elw((size_t)BT * N4 * 32), 256, 0, stream>>>(c0a, nei5, s0a, N5, N4, 32);
    float* aw0 = alloc((size_t)BT * N5 * 2);
    att_score<<<elw((size_t)BT * N5 * 32), 256, 0, stream>>>(c0f, c0a, F(A0_WF), F(A0_WA), aw0, N5, 32);
    float* f0 = alloc((size_t)BT * N5 * 32);
    float* a0 = alloc((size_t)BT * N5 * 32);
    launch_sconv(stream, c0f, nullptr, 1, F(A0_MF), nullptr, aw0, 0, f0, N5, 32, 32, 0);
    launch_sconv(stream, c0a, nullptr, 1, F(A0_MA), nullptr, aw0, 1, a0, N5, 32, 32, 0);

    // level 4
    float* c1f = alloc((size_t)BT * N4 * 64);
    float* c1a = alloc((size_t)BT * N4 * 64);
    launch_sconv(stream, s0f, nei4, 7, F(C1F_W), F(C1F_B), nullptr, 0, c1f, N4, 32, 64, 1);
    launch_sconv(stream, s0a, nei4, 7, F(C1A_W), F(C1A_B), nullptr, 0, c1a, N4, 32, 64, 1);
    float* s1f = alloc((size_t)BT * N3 * 64);
    float* s1a = alloc((size_t)BT * N3 * 64);
    spool_max<<<elw((size_t)BT * N3 * 64), 256, 0, stream>>>(c1f, nei4, s1f, N4, N3, 64);
    spool_max<<<elw((size_t)BT * N3 * 64), 256, 0, stream>>>(c1a, nei4, s1a, N4, N3, 64);
    float* aw1 = alloc((size_t)BT * N4 * 2);
    att_score<<<elw((size_t)BT * N4 * 32), 256, 0, stream>>>(c1f, c1a, F(A1_WF), F(A1_WA), aw1, N4, 64);
    float* f1 = alloc((size_t)BT * N4 * 64);
    float* a1 = alloc((size_t)BT * N4 * 64);
    launch_sconv(stream, c1f, nullptr, 1, F(A1_MF), nullptr, aw1, 0, f1, N4, 64, 64, 0);
    launch_sconv(stream, c1a, nullptr, 1, F(A1_MA), nullptr, aw1, 1, a1, N4, 64, 64, 0);

    // level 3
    float* c2f = alloc((size_t)BT * N3 * 128);
    float* c2a = alloc((size_t)BT * N3 * 128);
    launch_sconv(stream, s1f, nei3, 7, F(C2F_W), F(C2F_B), nullptr, 0, c2f, N3, 64, 128, 1);
    launch_sconv(stream, s1a, nei3, 7, F(C2A_W), F(C2A_B), nullptr, 0, c2a, N3, 64, 128, 1);
    float* s2f = alloc((size_t)BT * N2 * 128);
    float* s2a = alloc((size_t)BT * N2 * 128);
    spool_max<<<elw((size_t)BT * N2 * 128), 256, 0, stream>>>(c2f, nei3, s2f, N3, N2, 128);
    spool_max<<<elw((size_t)BT * N2 * 128), 256, 0, stream>>>(c2a, nei3, s2a, N3, N2, 128);
    float* aw2 = alloc((size_t)BT * N3 * 2);
    att_score<<<elw((size_t)BT * N3 * 32), 256, 0, stream>>>(c2f, c2a, F(A2_WF), F(A2_WA), aw2, N3, 128);
    float* f2 = alloc((size_t)BT * N3 * 128);
    float* a2 = alloc((size_t)BT * N3 * 128);
    launch_sconv(stream, c2f, nullptr, 1, F(A2_MF), nullptr, aw2, 0, f2, N3, 128, 128, 0);
    launch_sconv(stream, c2a, nullptr, 1, F(A2_MA), nullptr, aw2, 1, a2, N3, 128, 128, 0);

    // level 2
    float* c3f = alloc((size_t)BT * N2 * 256);
    float* c3a = alloc((size_t)BT * N2 * 256);
    launch_sconv(stream, s2f, nei2, 7, F(C3F_W), F(C3F_B), nullptr, 0, c3f, N2, 128, 256, 1);
    launch_sconv(stream, s2a, nei2, 7, F(C3A_W), F(C3A_B), nullptr, 0, c3a, N2, 128, 256, 1);
    float* s3f = alloc((size_t)BT * N1 * 256);
    float* s3a = alloc((size_t)BT * N1 * 256);
    spool_max<<<elw((size_t)BT * N1 * 256), 256, 0, stream>>>(c3f, nei2, s3f, N2, N1, 256);
    spool_max<<<elw((size_t)BT * N1 * 256), 256, 0, stream>>>(c3a, nei2, s3a, N2, N1, 256);
    float* aw3 = alloc((size_t)BT * N2 * 2);
    att_score<<<elw((size_t)BT * N2 * 32), 256, 0, stream>>>(c3f, c3a, F(A3_WF), F(A3_WA), aw3, N2, 256);
    float* f3 = alloc((size_t)BT * N2 * 256);
    float* a3 = alloc((size_t)BT * N2 * 256);
    launch_sconv(stream, c3f, nullptr, 1, F(A3_MF), nullptr, aw3, 0, f3, N2, 256, 256, 0);
    launch_sconv(stream, c3a, nullptr, 1, F(A3_MA), nullptr, aw3, 1, a3, N2, 256, 256, 0);

    // level 1
    float* c4f = alloc((size_t)BT * N1 * 512);
    float* c4a = alloc((size_t)BT * N1 * 512);
    launch_sconv(stream, s3f, nei1, 7, F(C4F_W), F(C4F_B), nullptr, 0, c4f, N1, 256, 512, 1);
    launch_sconv(stream, s3a, nei1, 7, F(C4A_W), F(C4A_B), nullptr, 0, c4a, N1, 256, 512, 1);
    float* aw4 = alloc((size_t)BT * N1 * 2);
    att_score<<<elw((size_t)BT * N1 * 32), 256, 0, stream>>>(c4f, c4a, F(A4_WF), F(A4_WA), aw4, N1, 512);
    float* f4 = alloc((size_t)BT * N1 * 256);
    float* a4 = alloc((size_t)BT * N1 * 256);
    launch_sconv(stream, c4f, nullptr, 1, F(A4_MF), nullptr, aw4, 0, f4, N1, 512, 256, 0);
    launch_sconv(stream, c4a, nullptr, 1, F(A4_MA), nullptr, aw4, 1, a4, N1, 512, 256, 0);

    // ---------- decoder ----------
    float* cat5 = alloc((size_t)BT * N2 * 1024);
    concat_gather<<<elw((size_t)BT * N2 * 1024), 256, 0, stream>>>(cat5, N2, 1024, up2,
        f4, 256, N1, 1,  a4, 256, N1, 1,  f3, 256, N2, 0,  a3, 256, N2, 0);
    float* o5 = alloc((size_t)BT * N2 * 256);
    launch_sconv(stream, cat5, nei2, 7, F(C5_W), F(C5_B), nullptr, 0, o5, N2, 1024, 256, 1);

    float* cat6 = alloc((size_t)BT * N3 * 512);
    concat_gather<<<elw((size_t)BT * N3 * 512), 256, 0, stream>>>(cat6, N3, 512, up3,
        o5, 256, N2, 1,  f2, 128, N3, 0,  a2, 128, N3, 0,  nullptr, 0, 1, 0);
    float* o6 = alloc((size_t)BT * N3 * 128);
    launch_sconv(stream, cat6, nei3, 7, F(C6_W), F(C6_B), nullptr, 0, o6, N3, 512, 128, 1);

    float* cat7 = alloc((size_t)BT * N4 * 256);
    concat_gather<<<elw((size_t)BT * N4 * 256), 256, 0, stream>>>(cat7, N4, 256, up4,
        o6, 128, N3, 1,  f1, 64, N4, 0,  a1, 64, N4, 0,  nullptr, 0, 1, 0);
    float* o7 = alloc((size_t)BT * N4 * 64);
    launch_sconv(stream, cat7, nei4, 7, F(C7_W), F(C7_B), nullptr, 0, o7, N4, 256, 64, 1);

    float* cat8 = alloc((size_t)BT * N5 * 128);
    concat_gather<<<elw((size_t)BT * N5 * 128), 256, 0, stream>>>(cat8, N5, 128, up5,
        o7, 64, N4, 1,  f0, 32, N5, 0,  a0, 32, N5, 0,  nullptr, 0, 1, 0);
    float* o8 = alloc((size_t)BT * N5 * 32);
    launch_sconv(stream, cat8, nei5, 7, F(C8_W), F(C8_B), nullptr, 0, o8, N5, 128, 32, 1);

    float* o9 = alloc((size_t)BT * N5);
    launch_sconv(stream, o8, nei5, 7, F(C9_W), F(C9_B), nullptr, 0, o9, N5, 32, 1, 0);

    softmax_rows<<<BT, 256, 0, stream>>>(o9, (float*)d_out, N5);
}